// Model_13606456393830
// MI455X (gfx1250) — compile-verified
//
#include <hip/hip_runtime.h>

typedef __attribute__((ext_vector_type(2))) float v2f;
typedef __attribute__((ext_vector_type(8))) float v8f;

#define FEAT 128
#define HID  256
#define LAB  40

__device__ __forceinline__ int imin(int a, int b) { return a < b ? a : b; }

// ---------------- utility: zero a float buffer ----------------
__global__ __launch_bounds__(256) void zero_kernel(float* __restrict__ p, size_t n) {
  size_t i = (size_t)blockIdx.x * 256 + threadIdx.x;
  if (i < n) p[i] = 0.0f;
}

// ---------------- degree accumulation ----------------
__global__ __launch_bounds__(256) void degree_kernel(const int* __restrict__ src,
                                                     const int* __restrict__ dst,
                                                     float* __restrict__ outdeg,
                                                     float* __restrict__ indeg,
                                                     int n_edges) {
  int i = blockIdx.x * 256 + threadIdx.x;
  if (i < n_edges) {
    atomicAdd(&outdeg[src[i]], 1.0f);
    atomicAdd(&indeg[dst[i]], 1.0f);
  }
}

// c[i] = rsqrt(max(deg,1)) in place over both deg arrays (contiguous)
__global__ __launch_bounds__(256) void coeff_kernel(float* __restrict__ c, int n2) {
  int i = blockIdx.x * 256 + threadIdx.x;
  if (i < n2) c[i] = rsqrtf(fmaxf(c[i], 1.0f));
}

// ---------------- edge scatter: agg[dst] += h[src] * c_src[src] ----------------
// One wave (32 lanes) per edge; float4 gathers, f32 global atomics for the scatter.
template <int F>
__global__ __launch_bounds__(256) void scatter_edges(const float* __restrict__ H,
                                                     const int* __restrict__ src,
                                                     const int* __restrict__ dst,
                                                     const float* __restrict__ csrc,
                                                     float* __restrict__ agg,
                                                     int n_edges) {
  int wid  = (int)((blockIdx.x * 256 + threadIdx.x) >> 5);
  int lane = threadIdx.x & 31;
  if (wid >= n_edges) return;
  int s = src[wid];
  int d = dst[wid];
  float cs = csrc[s];
  const float* __restrict__ hp = H + (size_t)s * F;
  float* __restrict__ op = agg + (size_t)d * F;
#pragma unroll
  for (int f = lane * 4; f < F; f += 128) {
    float4 v = *reinterpret_cast<const float4*>(hp + f);
    atomicAdd(op + f + 0, v.x * cs);
    atomicAdd(op + f + 1, v.y * cs);
    atomicAdd(op + f + 2, v.z * cs);
    atomicAdd(op + f + 3, v.w * cs);
  }
}

// ---------------- WMMA f32 GEMM: out = act(rowscale ⊙ (A @ W^T) + bias) ----------------
// A: [n_rows, FIN] row-major. W: [fout, FIN] row-major (torch Linear layout).
// One wave computes a 16-row strip across NTILES 16-col tiles using
// V_WMMA_F32_16X16X4_F32. Load indices are CLAMPED (not predicated): garbage in
// A row r / B col n only pollutes C row r / col n, which the epilogue never
// stores. This keeps the K-loop 100% branch-free (EXEC untouched).
template <int FIN, int NTILES>
__global__ __launch_bounds__(256) void gemm_wmma_f32(const float* __restrict__ A,
                                                     const float* __restrict__ W,
                                                     const float* __restrict__ rowscale,
                                                     const float* __restrict__ bias,
                                                     float* __restrict__ out,
                                                     int n_rows, int fout, int do_relu) {
  const int lane = threadIdx.x & 31;
  const int wave = threadIdx.x >> 5;
  const int m0 = (blockIdx.x * 8 + wave) * 16;
  if (m0 >= n_rows) return;  // wave-uniform exit

  const int koff = (lane >> 4) * 2;   // K sub-offset per lane half
  const int colW = lane & 15;         // per-tile column offset

  // Clamped fragment base pointers (always in-bounds).
  const int rowA = imin(m0 + (lane & 15), n_rows - 1);
  const float* __restrict__ ap = A + (size_t)rowA * FIN + koff;
  const float* bp[NTILES];
#pragma unroll
  for (int t = 0; t < NTILES; ++t) {
    int n = imin(t * 16 + colW, fout - 1);
    bp[t] = W + (size_t)n * FIN + koff;
  }

  v8f acc[NTILES];
  const v8f vzero = {};
#pragma unroll
  for (int t = 0; t < NTILES; ++t) acc[t] = vzero;

#pragma unroll 2
  for (int k0 = 0; k0 < FIN; k0 += 4) {
    // Batch all fragment loads, then the WMMA chain (loadcnt drains gradually).
    v2f a = *reinterpret_cast<const v2f*>(ap + k0);
    v2f bf[NTILES];
#pragma unroll
    for (int t = 0; t < NTILES; ++t)
      bf[t] = *reinterpret_cast<const v2f*>(bp[t] + k0);
#pragma unroll
    for (int t = 0; t < NTILES; ++t)
      acc[t] = __builtin_amdgcn_wmma_f32_16x16x4_f32(
          /*neg_a=*/false, a, /*neg_b=*/false, bf[t],
          /*c_mod=*/(short)0, acc[t], /*reuse_a=*/false, /*reuse_b=*/false);
  }

  // Epilogue: C layout -> VGPR r: lanes0-15 M=r, lanes16-31 M=r+8; N = lane&15.
  const int rbase = (lane >> 4) * 8;
#pragma unroll
  for (int t = 0; t < NTILES; ++t) {
    int n = t * 16 + colW;
    if (n < fout) {
      float bn = bias[n];
#pragma unroll
      for (int r = 0; r < 8; ++r) {
        int m = m0 + rbase + r;
        if (m < n_rows) {
          float v = acc[t][r] * rowscale[m] + bn;
          if (do_relu) v = fmaxf(v, 0.0f);
          out[(size_t)m * fout + n] = v;
        }
      }
    }
  }
}

extern "C" void kernel_launch(void* const* d_in, const int* in_sizes, int n_in,
                              void* d_out, int out_size, void* d_ws, size_t ws_size,
                              hipStream_t stream) {
  const float* feats = (const float*)d_in[0];
  const int*   src   = (const int*)d_in[1];
  const int*   dst   = (const int*)d_in[2];
  const float* W1    = (const float*)d_in[3];
  const float* b1    = (const float*)d_in[4];
  const float* W2    = (const float*)d_in[5];
  const float* b2    = (const float*)d_in[6];
  const float* W3    = (const float*)d_in[7];
  const float* b3    = (const float*)d_in[8];
  float*       out   = (float*)d_out;

  const int n = in_sizes[0] / FEAT;   // 100000
  const int E = in_sizes[1];          // 1600000

  float* ws   = (float*)d_ws;
  float* csrc = ws;                         // n
  float* cdst = ws + n;                     // n
  float* bufA = ws + 2 * (size_t)n;         // n*256 (agg)
  float* bufB = bufA + (size_t)n * HID;     // n*256 (activations)

  dim3 blk(256);
  auto blocksFor = [](size_t elems) { return (unsigned)((elems + 255) / 256); };

  // --- degree coefficients ---
  zero_kernel<<<blocksFor(2 * (size_t)n), blk, 0, stream>>>(csrc, 2 * (size_t)n);
  degree_kernel<<<blocksFor(E), blk, 0, stream>>>(src, dst, csrc, cdst, E);
  coeff_kernel<<<blocksFor(2 * (size_t)n), blk, 0, stream>>>(csrc, 2 * n);

  const unsigned edgeBlocks = (unsigned)((E + 7) / 8);     // 8 waves/block, 1 edge/wave
  const unsigned gemmBlocks = (unsigned)((n + 127) / 128); // 8 waves/block, 16 rows/wave

  // --- layer 1: feats[128] -> hidden[256], relu ---
  zero_kernel<<<blocksFor((size_t)n * FEAT), blk, 0, stream>>>(bufA, (size_t)n * FEAT);
  scatter_edges<FEAT><<<edgeBlocks, blk, 0, stream>>>(feats, src, dst, csrc, bufA, E);
  gemm_wmma_f32<FEAT, 16><<<gemmBlocks, blk, 0, stream>>>(bufA, W1, cdst, b1, bufB, n, HID, 1);

  // --- layer 2: hidden[256] -> hidden[256], relu ---
  zero_kernel<<<blocksFor((size_t)n * HID), blk, 0, stream>>>(bufA, (size_t)n * HID);
  scatter_edges<HID><<<edgeBlocks, blk, 0, stream>>>(bufB, src, dst, csrc, bufA, E);
  gemm_wmma_f32<HID, 16><<<gemmBlocks, blk, 0, stream>>>(bufA, W2, cdst, b2, bufB, n, HID, 1);

  // --- layer 3: hidden[256] -> labels[40], no activation ---
  zero_kernel<<<blocksFor((size_t)n * HID), blk, 0, stream>>>(bufA, (size_t)n * HID);
  scatter_edges<HID><<<edgeBlocks, blk, 0, stream>>>(bufB, src, dst, csrc, bufA, E);
  gemm_wmma_f32<HID, 3><<<gemmBlocks, blk, 0, stream>>>(bufA, W3, cdst, b3, out, n, LAB, 0);
}